// DepthwiseConv3D_68195490725926
// MI455X (gfx1250) — compile-verified
//
#include <hip/hip_runtime.h>
#include <stdint.h>

// ---------------- problem constants (from reference) ----------------
namespace {
constexpr int kB = 4, kD = 16, kH = 112, kW = 112, kC = 64;
constexpr int kDo = 14, kHo = 110, kWo = 110;

constexpr int HT = 8;                      // output rows per block (== waves)
constexpr int WT = 16;                     // output cols per block
constexpr int HH = HT + 2, WW = WT + 2;    // input halo tile 10 x 18
constexpr int SLICE_F  = HH * WW * kC;     // 11520 floats per depth slice
constexpr int SLICE_V4 = SLICE_F / 4;      // 2880 16-byte chunks
constexpr int ROW_V4   = WW * kC / 4;      // 288 chunks per halo row
constexpr int NBUF     = 3;                // rolling depth-slice window
constexpr int THREADS  = 256;              // 8 wave32 waves
constexpr int NPER     = (SLICE_V4 + THREADS - 1) / THREADS; // 12
constexpr int HTILES = (kHo + HT - 1) / HT;  // 14
constexpr int WTILES = (kWo + WT - 1) / WT;  // 7
constexpr int WCHUNKS = 27 * kC / 4;         // 432 weight 16B chunks
}

// ---------------- gfx1250 async global->LDS staging ----------------
#if __has_builtin(__builtin_amdgcn_global_load_async_to_lds_b128)
#define HAVE_ASYNC_LDS 1
#else
#define HAVE_ASYNC_LDS 0
#warning "gfx1250 async-to-LDS builtin not found: using synchronous staging fallback"
#endif

// Builtin signature (probed): (v4i addrspace(1)*, v4i addrspace(3)*, imm, imm)
typedef int v4i_t __attribute__((vector_size(16)));
typedef __attribute__((address_space(1))) v4i_t* gas_v4i;
typedef __attribute__((address_space(3))) v4i_t* las_v4i;

// native 2-float vector for nontemporal stores (builtin rejects HIP_vector_type)
typedef float v2f_t __attribute__((ext_vector_type(2)));

__device__ __forceinline__ void stage16(const float* g, float* l) {
#if HAVE_ASYNC_LDS
  // AS(3) pointers are 32-bit LDS byte offsets == low 32 bits of the
  // generic shared pointer, so truncation is the correct conversion.
  __builtin_amdgcn_global_load_async_to_lds_b128(
      (gas_v4i)(uintptr_t)g,
      (las_v4i)(uint32_t)(uintptr_t)l,
      /*offset=*/0, /*cpol=*/0);
#else
  *(float4*)l = *(const float4*)g;
#endif
}

__device__ __forceinline__ void async_drain() {
#if HAVE_ASYNC_LDS
#if __has_builtin(__builtin_amdgcn_s_wait_asynccnt)
  __builtin_amdgcn_s_wait_asynccnt(0);
#else
  asm volatile("s_wait_asynccnt 0" ::: "memory");
#endif
#endif
}

// ---------------- kernel ----------------
__global__ __launch_bounds__(THREADS)
void DepthwiseConv3D_68195490725926_kernel(const float* __restrict__ x,
                                           const float* __restrict__ w,
                                           const float* __restrict__ bias,
                                           float* __restrict__ out) {
  __shared__ float lds_x[NBUF * SLICE_F];  // 138,240 B
  __shared__ float lds_w[27 * kC];         //   6,912 B

  const int tid  = threadIdx.x;
  const int lane = tid & 31;   // channel pair: channels 2*lane, 2*lane+1
  const int wave = tid >> 5;   // output row within tile

  const int wt = blockIdx.x % WTILES;
  const int ht = (blockIdx.x / WTILES) % HTILES;
  const int b  = blockIdx.x / (WTILES * HTILES);

  const int h0 = ht * HT;
  const int w0 = wt * WT;

  const float* xb = x + b * (kD * kH * kW * kC);

  // ---- stage weights (27*64 floats) into LDS (async) ----
  {
    int e = tid;
    if (e < WCHUNKS) stage16(w + e * 4, lds_w + e * 4);
    e = tid + THREADS;
    if (e < WCHUNKS) stage16(w + e * 4, lds_w + e * 4);
  }

  // per-lane bias (2 channels)
  const float2 bv = *(const float2*)(bias + 2 * lane);

  // ---- precompute depth-invariant staging offsets (12 chunks/thread) ----
  int goff[NPER];   // float offset within one input depth slice
  int loff[NPER];   // float offset within one LDS slice slot
#pragma unroll
  for (int i = 0; i < NPER; ++i) {
    int e = tid + i * THREADS;
    if (e >= SLICE_V4) e = SLICE_V4 - 1;        // benign duplicate write
    const int row = e / ROW_V4;                 // 0..9
    const int fo  = (e - row * ROW_V4) * 4;     // float offset in halo row
    int h_in = h0 + row;       if (h_in > kH - 1) h_in = kH - 1;  // clamps feed
    int w_in = w0 + (fo >> 6); if (w_in > kW - 1) w_in = kW - 1;  // discarded outs only
    goff[i] = (h_in * kW + w_in) * kC + (fo & 63);
    loff[i] = row * (WW * kC) + fo;
  }

  auto stage_slice = [&](int d_in, int slot) {
    const float* xd = xb + d_in * (kH * kW * kC);
    float* lb = lds_x + slot * SLICE_F;
#pragma unroll
    for (int i = 0; i < NPER; ++i) stage16(xd + goff[i], lb + loff[i]);
  };

  // one kd-plane of taps: acc[j] += sum_kh sum_kw x * w
  auto accum_kd = [&](int kd, int slot, float2* acc) {
    const float* sl = lds_x + slot * SLICE_F;
#pragma unroll
    for (int kh = 0; kh < 3; ++kh) {
      const float* srow = sl + (wave + kh) * (WW * kC) + 2 * lane;
      const float* wr   = lds_w + (kd * 3 + kh) * 3 * kC + 2 * lane;
      const float2 w0v = *(const float2*)(wr);
      const float2 w1v = *(const float2*)(wr + kC);
      const float2 w2v = *(const float2*)(wr + 2 * kC);
      float2 xin[WW];
#pragma unroll
      for (int u = 0; u < WW; ++u) xin[u] = *(const float2*)(srow + u * kC);
#pragma unroll
      for (int j = 0; j < WT; ++j) {   // sliding-window reuse along W
        acc[j].x = fmaf(xin[j    ].x, w0v.x, acc[j].x);
        acc[j].y = fmaf(xin[j    ].y, w0v.y, acc[j].y);
        acc[j].x = fmaf(xin[j + 1].x, w1v.x, acc[j].x);
        acc[j].y = fmaf(xin[j + 1].y, w1v.y, acc[j].y);
        acc[j].x = fmaf(xin[j + 2].x, w2v.x, acc[j].x);
        acc[j].y = fmaf(xin[j + 2].y, w2v.y, acc[j].y);
      }
    }
  };

  // ---- prologue: slices 0,1 + weights resident before first partial compute
  stage_slice(0, 0);
  stage_slice(1, 1);
  async_drain();
  __syncthreads();

  for (int d_out = 0; d_out < kDo; ++d_out) {
    // issue DMA for slice d_out+2; its slot held slice d_out-1, which no wave
    // reads after the previous iteration's trailing barrier.
    stage_slice(d_out + 2, (d_out + 2) % NBUF);

    float2 acc[WT];
#pragma unroll
    for (int j = 0; j < WT; ++j) acc[j] = bv;

    // overlap the DMA with the 2/3 of the stencil that uses resident slices
    accum_kd(0, d_out % NBUF, acc);
    accum_kd(1, (d_out + 1) % NBUF, acc);

    async_drain();        // slice d_out+2 landed
    __syncthreads();      // ...in every wave

    accum_kd(2, (d_out + 2) % NBUF, acc);

    // store this depth plane's row (guard right/bottom edges); streaming NT
    const int h_out = h0 + wave;
    if (h_out < kHo) {
      float* op = out + (((b * kDo + d_out) * kHo + h_out) * kWo) * kC + 2 * lane;
#pragma unroll
      for (int j = 0; j < WT; ++j) {
        const int w_out = w0 + j;
        if (w_out < kWo) {
          v2f_t v;
          v.x = acc[j].x;
          v.y = acc[j].y;
          __builtin_nontemporal_store(v, (v2f_t*)(op + w_out * kC));
        }
      }
    }
    __syncthreads();  // protect LDS slot before next iteration's DMA issue
  }
}

// ---------------- host launcher ----------------
extern "C" void kernel_launch(void* const* d_in, const int* in_sizes, int n_in,
                              void* d_out, int out_size, void* d_ws, size_t ws_size,
                              hipStream_t stream) {
  (void)in_sizes; (void)n_in; (void)out_size; (void)d_ws; (void)ws_size;
  const float* x    = (const float*)d_in[0];
  const float* w    = (const float*)d_in[1];
  const float* bias = (const float*)d_in[2];
  float* out = (float*)d_out;

  dim3 grid(kB * HTILES * WTILES);  // 392 blocks
  DepthwiseConv3D_68195490725926_kernel<<<grid, THREADS, 0, stream>>>(x, w, bias, out);
}